// GINLayer_36369783062754
// MI455X (gfx1250) — compile-verified
//
#include <hip/hip_runtime.h>
#include <hip/hip_bf16.h>

typedef __attribute__((ext_vector_type(2))) float v2f;
typedef __attribute__((ext_vector_type(8))) float v8f;

#define D_FEAT   128
#define WSTRIDE  136   // 136%64==8 -> B-frag reads conflict-free; 16B-aligned rows
#define ASTRIDE  132   // 132%64==4 -> A-frag reads conflict-free; 16B-aligned rows

// ---------------------------------------------------------------------------
// CDNA5 async global->LDS DMA (no VGPR round-trip), tracked by ASYNCcnt.
// lds_byte_addr: low 32 bits of generic __shared__ pointer == LDS offset.
// ---------------------------------------------------------------------------
__device__ __forceinline__ void async_copy_b128(unsigned lds_byte_addr, const void* gptr) {
    asm volatile("global_load_async_to_lds_b128 %0, %1, off"
                 :
                 : "v"(lds_byte_addr), "v"(gptr)
                 : "memory");
}
__device__ __forceinline__ void wait_async_zero() {
    asm volatile("s_wait_asynccnt 0" ::: "memory");
}

// ---------------------------------------------------------------------------
// SpMM scatter: combined[row] += val * features[col]   (one wave32 per edge)
// Edge metadata is wave-uniform -> readfirstlane promotes to SMEM loads.
// Each lane owns a float4 of the 128-wide feature row -> 4 fp32 HW atomics.
// features/combined are L2-resident (51 MB each vs 192 MB L2).
// ---------------------------------------------------------------------------
__launch_bounds__(256)
__global__ void spmm_scatter_kernel(const int* __restrict__ rows,
                                    const int* __restrict__ cols,
                                    const float* __restrict__ vals,
                                    const float* __restrict__ feat,
                                    float* __restrict__ combined,
                                    int nEdges)
{
    int gtid = blockIdx.x * blockDim.x + threadIdx.x;
    int edge = __builtin_amdgcn_readfirstlane(gtid >> 5);
    int lane = threadIdx.x & 31;
    if (edge >= nEdges) return;

    int   r = rows[edge];   // uniform -> s_load
    int   c = cols[edge];   // uniform -> s_load
    float v = vals[edge];   // uniform -> s_load

    const float4 f = ((const float4*)(feat + (size_t)c * D_FEAT))[lane];
    float* dst = combined + (size_t)r * D_FEAT + lane * 4;

    unsafeAtomicAdd(dst + 0, v * f.x);
    unsafeAtomicAdd(dst + 1, v * f.y);
    unsafeAtomicAdd(dst + 2, v * f.z);
    unsafeAtomicAdd(dst + 3, v * f.w);
}

// ---------------------------------------------------------------------------
// D[M,128] = act(A[M,128] @ W[128,128] + bias)  using V_WMMA_F32_16X16X4_F32.
// 256 threads = 8 waves; each wave computes a 16x128 output stripe:
//   8 accumulators (16x16 f32, v8f each, bias pre-splatted), 32 k-steps of 4.
// W and per-wave A tiles staged in LDS via async global->LDS DMA.
//
// WMMA fragment layouts (ISA 7.12.2, 32-bit):
//   A 16x4 : lane l, vgpr v -> M = l%16, K = (l/16)*2 + v
//   B 4x16 : lane l, vgpr v -> N = l%16, K = (l/16)*2 + v
//   C/D    : lane l, vgpr v -> N = l%16, M = v + (l/16)*8
// ---------------------------------------------------------------------------
__launch_bounds__(256)
__global__ void gemm128_wmma_kernel(const float* __restrict__ A,
                                    const float* __restrict__ W,
                                    const float* __restrict__ bias,
                                    float* __restrict__ D,
                                    int M, int do_relu)
{
    __shared__ float sW[128 * WSTRIDE];
    __shared__ float sA[8][16 * ASTRIDE];

    const int tid  = threadIdx.x;
    const int wave = tid >> 5;
    const int lane = tid & 31;
    const int half = lane >> 4;   // 0 or 1
    const int l16  = lane & 15;

    // Stage W (128x128) into LDS: async DMA, 16B per lane per issue.
    for (int i = tid; i < 128 * 128 / 4; i += 256) {
        int r = (i * 4) >> 7;
        int c = (i * 4) & 127;
        unsigned lds = (unsigned)(uintptr_t)&sW[r * WSTRIDE + c];
        async_copy_b128(lds, (const void*)(W + i * 4));
    }

    const int row0    = (blockIdx.x * 8 + wave) * 16;
    const bool active = row0 < M;

    // Stage this wave's 16x128 A tile into LDS via async DMA.
    if (active) {
        const float* Ag = A + (size_t)row0 * D_FEAT;
        for (int i = lane; i < 16 * 32; i += 32) {  // 16 rows x 32 float4
            int r  = i >> 5;
            int c4 = i & 31;
            unsigned lds = (unsigned)(uintptr_t)&sA[wave][r * ASTRIDE + c4 * 4];
            async_copy_b128(lds, (const void*)(Ag + i * 4));
        }
    }

    wait_async_zero();
    __syncthreads();

    if (!active) return;

    // Accumulators pre-loaded with bias (C/D element (M,N) gets bias[N]).
    v8f acc[8];
#pragma unroll
    for (int nt = 0; nt < 8; nt++) {
        float b = bias[nt * 16 + l16];
#pragma unroll
        for (int v = 0; v < 8; v++) acc[nt][v] = b;
    }

    const float* Arow = &sA[wave][l16 * ASTRIDE];
    for (int k = 0; k < 128; k += 4) {
        v2f afrag;
        afrag.x = Arow[k + half * 2 + 0];
        afrag.y = Arow[k + half * 2 + 1];
#pragma unroll
        for (int nt = 0; nt < 8; nt++) {
            v2f bfrag;
            const float* Wp = &sW[(k + half * 2) * WSTRIDE + nt * 16 + l16];
            bfrag.x = Wp[0];
            bfrag.y = Wp[WSTRIDE];
            acc[nt] = __builtin_amdgcn_wmma_f32_16x16x4_f32(
                false, afrag, false, bfrag, (short)0, acc[nt], false, false);
        }
    }

    // Epilogue: optional ReLU and store. Lanes 0-15 of a VGPR cover 16
    // consecutive columns of one row -> 64B contiguous per half-wave.
#pragma unroll
    for (int nt = 0; nt < 8; nt++) {
        int n = nt * 16 + l16;
#pragma unroll
        for (int v = 0; v < 8; v++) {
            float val = acc[nt][v];
            if (do_relu) val = fmaxf(val, 0.0f);
            D[(size_t)(row0 + v + half * 8) * D_FEAT + n] = val;
        }
    }
}

extern "C" void kernel_launch(void* const* d_in, const int* in_sizes, int n_in,
                              void* d_out, int out_size, void* d_ws, size_t ws_size,
                              hipStream_t stream) {
    const int*   indices  = (const int*)d_in[0];   // [2, E] row-major
    const float* values   = (const float*)d_in[1]; // [E]
    const float* features = (const float*)d_in[2]; // [N, 128]
    const float* W1       = (const float*)d_in[3];
    const float* b1       = (const float*)d_in[4];
    const float* W2       = (const float*)d_in[5];
    const float* b2       = (const float*)d_in[6];
    float*       out      = (float*)d_out;

    const int E = in_sizes[0] / 2;
    const int N = in_sizes[2] / D_FEAT;

    float* combined = (float*)d_ws;                       // [N,128]
    float* h        = combined + (size_t)N * D_FEAT;      // [N,128]

    // combined = (1+eps)*features, eps == 0 -> plain copy (D2D, capturable)
    hipMemcpyAsync(combined, features, (size_t)N * D_FEAT * sizeof(float),
                   hipMemcpyDeviceToDevice, stream);

    // combined += segment_sum(values * features[col])
    int spmmBlocks = (E + 7) / 8;  // 8 edges (waves) per 256-thread block
    spmm_scatter_kernel<<<spmmBlocks, 256, 0, stream>>>(
        indices, indices + E, values, features, combined, E);

    // h = relu(combined @ W1 + b1); out = h @ W2 + b2
    int mtiles = (N + 15) / 16;
    int gemmBlocks = (mtiles + 7) / 8;
    gemm128_wmma_kernel<<<gemmBlocks, 256, 0, stream>>>(combined, W1, b1, h, N, 1);
    gemm128_wmma_kernel<<<gemmBlocks, 256, 0, stream>>>(h, W2, b2, out, N, 0);
}